// BilateralGrid_77421080478221
// MI455X (gfx1250) — compile-verified
//
#include <hip/hip_runtime.h>

// Bilateral grid slicing + per-pixel affine apply for MI455X (gfx1250).
// - 96KB coefficient grid staged to LDS via CDNA5 async global->LDS loads
//   (ASYNCcnt path), amortized over a grid-stride pixel loop.
// - Trilinear interp as 8-corner weighted sum (v_pk_fma_f32-friendly),
//   single LDS base address + immediate ds offsets for all 8 corners.

#define AS1 __attribute__((address_space(1)))
#define AS3 __attribute__((address_space(3)))

typedef __attribute__((ext_vector_type(4))) int   v4i;
typedef __attribute__((ext_vector_type(4))) float v4f;

__global__ __launch_bounds__(256) void bilateral_grid_kernel(
    const float* __restrict__ pixels,   // (N,3)
    const float* __restrict__ coords,   // (N,2)
    const float* __restrict__ grid,     // (16,16,8,12)
    float* __restrict__ out,            // (N,3)
    int N)
{
    extern __shared__ float sgrid[];    // 16*16*8*12 = 24576 floats = 96KB

    // ---- Stage grid into LDS with async global->LDS (ASYNCcnt path) ----
    constexpr int NVEC = (16 * 16 * 8 * 12) / 4;   // 6144 x 16B transfers
    for (int v = threadIdx.x; v < NVEC; v += blockDim.x) {
        __builtin_amdgcn_global_load_async_to_lds_b128(
            (AS1 v4i*)(grid + 4 * v),
            (AS3 v4i*)(sgrid + 4 * v),
            /*offset=*/0, /*cpol=*/0);
    }
    __builtin_amdgcn_s_wait_asynccnt(0);
    __syncthreads();

    int i = blockIdx.x * blockDim.x + threadIdx.x;
    const int stride = gridDim.x * blockDim.x;

    for (; i < N; i += stride) {
        // Streaming loads (b96 / b64, coalesced across lanes)
        const float r = pixels[3 * i + 0];
        const float g = pixels[3 * i + 1];
        const float b = pixels[3 * i + 2];
        const float cx = coords[2 * i + 0];
        const float cy = coords[2 * i + 1];

        // Guide = luminance
        const float guide = fmaf(0.2126f, r, fmaf(0.7152f, g, 0.0722f * b));

        // Grid coordinates (Sh=16, Sw=16, Sl=8); clamps guarantee
        // x0<=14, y0<=14, z0<=6, so +1 neighbors are always in range.
        const float gx = fminf(fmaxf(cx * 15.0f, 0.0f), 14.999f);
        const float gy = fminf(fmaxf(cy * 15.0f, 0.0f), 14.999f);
        const float gcl = fminf(fmaxf(guide, 0.0f), 1.0f);
        const float gz = fminf(fmaxf(gcl * 7.0f, 0.0f), 6.999f);

        const int x0 = (int)gx;            // floor (non-negative)
        const int y0 = (int)gy;
        const int z0 = (int)gz;
        const float fx = gx - (float)x0;
        const float fy = gy - (float)y0;
        const float fz = gz - (float)z0;

        // Trilinear weights (8 corners), labeled w[y][x][z]
        const float u0 = 1.0f - fx, u1 = fx;
        const float v0 = 1.0f - fy, v1 = fy;
        const float t0 = 1.0f - fz, t1 = fz;
        const float w00 = v0 * u0, w01 = v0 * u1;
        const float w10 = v1 * u0, w11 = v1 * u1;
        const float w000 = w00 * t0, w001 = w00 * t1;
        const float w010 = w01 * t0, w011 = w01 * t1;
        const float w100 = w10 * t0, w101 = w10 * t1;
        const float w110 = w11 * t0, w111 = w11 * t1;

        // Single LDS base; all 8 corners are constant float4 offsets:
        // z+1 -> +3, x+1 -> +24, y+1 -> +384 (float4 units; all <64KB bytes)
        const int cell = (y0 * 128 + x0 * 8 + z0) * 3;     // float4 units
        const v4f* s4b = (const v4f*)sgrid + cell;

        v4f coef[3];
#pragma unroll
        for (int k = 0; k < 3; ++k) {
            v4f a;
            a  = s4b[0   + k] * w000;   // (y0,x0,z0)
            a += s4b[3   + k] * w001;   // (y0,x0,z1)
            a += s4b[24  + k] * w010;   // (y0,x1,z0)
            a += s4b[27  + k] * w011;   // (y0,x1,z1)
            a += s4b[384 + k] * w100;   // (y1,x0,z0)
            a += s4b[387 + k] * w101;   // (y1,x0,z1)
            a += s4b[408 + k] * w110;   // (y1,x1,z0)
            a += s4b[411 + k] * w111;   // (y1,x1,z1)
            coef[k] = a;
        }

        // Affine apply: out = A[:, :3] @ rgb + A[:, 3]
        const float o0 = fmaf(coef[0].x, r, fmaf(coef[0].y, g, fmaf(coef[0].z, b, coef[0].w)));
        const float o1 = fmaf(coef[1].x, r, fmaf(coef[1].y, g, fmaf(coef[1].z, b, coef[1].w)));
        const float o2 = fmaf(coef[2].x, r, fmaf(coef[2].y, g, fmaf(coef[2].z, b, coef[2].w)));

        out[3 * i + 0] = o0;
        out[3 * i + 1] = o1;
        out[3 * i + 2] = o2;
    }
}

extern "C" void kernel_launch(void* const* d_in, const int* in_sizes, int n_in,
                              void* d_out, int out_size, void* d_ws, size_t ws_size,
                              hipStream_t stream) {
    (void)n_in; (void)d_ws; (void)ws_size; (void)out_size;

    const float* pixels = (const float*)d_in[0];   // (B,H,W,3) f32
    const float* coords = (const float*)d_in[1];   // (B,H,W,2) f32
    const float* grid   = (const float*)d_in[2];   // (16,16,8,12) f32
    float* out = (float*)d_out;

    const int N = in_sizes[0] / 3;                 // number of pixels

    constexpr int kBlock = 256;
    constexpr size_t kLds = 16 * 16 * 8 * 12 * sizeof(float);  // 96KB

    // Allow >64KB dynamic LDS (gfx1250 has 320KB/WGP). Not a stream op;
    // safe under graph capture and deterministic.
    static_assert(kLds == 98304, "grid LDS size");
    (void)hipFuncSetAttribute((const void*)bilateral_grid_kernel,
                              hipFuncAttributeMaxDynamicSharedMemorySize, (int)kLds);

    int blocks = (N + kBlock - 1) / kBlock;
    if (blocks > 1536) blocks = 1536;   // cap re-staging traffic; grid-stride covers rest

    bilateral_grid_kernel<<<blocks, kBlock, kLds, stream>>>(pixels, coords, grid, out, N);
}